// HexDepthwiseConv_43894565765175
// MI455X (gfx1250) — compile-verified
//
#include <hip/hip_runtime.h>

#define N_NODES 50000
#define N_EDGES (7 * N_NODES)
#define BATCH 4
#define DIM 128
#define N_TYPES 7

typedef float v4f __attribute__((ext_vector_type(4)));

// ---------------------------------------------------------------------------
// Edge scatter.  One wave32 per edge (grid-stride); lane l owns dims
// [4l, 4l+4).  Weight table staged to LDS via CDNA5 async global->LDS copy
// (ASYNCcnt); edge indices read on the scalar (SMEM/KMcnt) path so the vector
// memory pipe is dedicated to the x-gather + f32 atomic-add stream (L2-
// resident: out = 102 MB < 192 MB L2).
// ---------------------------------------------------------------------------
__global__ __launch_bounds__(256) void hex_edge_scatter(const float* __restrict__ x,
                                                        const int* __restrict__ src,
                                                        const int* __restrict__ dst,
                                                        const int* __restrict__ typ,
                                                        const float* __restrict__ weight,
                                                        float* __restrict__ out,
                                                        int n_edges) {
  __shared__ v4f wlds[N_TYPES * (DIM / 4)];   // 7 * 32 float4 = 3584 B

  const int tid = threadIdx.x;

  // --- Stage weight[1,7,128] into LDS with async global->LDS b128 copies. ---
  if (tid < N_TYPES * (DIM / 4)) {
    // Low 32 bits of a generic pointer to __shared__ == LDS byte offset
    // (flat-aperture mapping discards the upper bits).
    unsigned lds_off = (unsigned)(size_t)(&wlds[tid]);
    const v4f* gsrc = ((const v4f*)weight) + tid;
    asm volatile("global_load_async_to_lds_b128 %0, %1, off"
                 :: "v"(lds_off), "v"(gsrc)
                 : "memory");
  }
  asm volatile("s_wait_asynccnt 0x0" ::: "memory");
  __syncthreads();

  const int lane = tid & 31;
  // Force the wave index to be compiler-visibly uniform so the per-edge index
  // loads below have uniform addresses -> scalar s_load (KMcnt) path.
  const int wave   = __builtin_amdgcn_readfirstlane(tid >> 5);
  const int wpb    = (int)(blockDim.x >> 5);
  const int gwave  = __builtin_amdgcn_readfirstlane(blockIdx.x) * wpb + wave;
  const int nwaves = (int)gridDim.x * wpb;

  for (int e = gwave; e < n_edges; e += nwaves) {
    // Prefetch next grid-stride iteration's edge records (sequential streams).
    int en = e + nwaves;
    if (en < n_edges) {
      __builtin_prefetch(&src[en], 0, 0);
      __builtin_prefetch(&dst[en], 0, 0);
      __builtin_prefetch(&typ[en], 0, 0);
    }

    const int s = src[e];   // uniform address -> scalar load
    const int d = dst[e];
    const int t = typ[e];

    const v4f wv = wlds[t * (DIM / 4) + lane];

    const v4f* xb = (const v4f*)x + (size_t)s * (DIM / 4) + lane;
    float*     ob = out + (size_t)d * DIM + lane * 4;

#pragma unroll
    for (int b = 0; b < BATCH; ++b) {
      v4f xv = xb[(size_t)b * N_NODES * (DIM / 4)];
      v4f m  = xv * wv;
      float* o = ob + (size_t)b * N_NODES * DIM;
      atomicAdd(o + 0, m.x);
      atomicAdd(o + 1, m.y);
      atomicAdd(o + 2, m.z);
      atomicAdd(o + 3, m.w);
    }
  }
}

// ---------------------------------------------------------------------------
// out[b,n,:] = bias[:]   (float4-vectorized; folds the +bias into the init)
// ---------------------------------------------------------------------------
__global__ __launch_bounds__(256) void hex_init_out(float* __restrict__ out,
                                                    const float* __restrict__ bias,
                                                    int n4) {
  int i = blockIdx.x * blockDim.x + threadIdx.x;
  // DIM=128 floats -> 32 float4 per row; flat float4 index mod 32 == dim chunk.
  v4f bv = ((const v4f*)bias)[i & 31];
  if (i < n4) ((v4f*)out)[i] = bv;
}

// ---------------------------------------------------------------------------
// Launch
// ---------------------------------------------------------------------------
extern "C" void kernel_launch(void* const* d_in, const int* in_sizes, int n_in,
                              void* d_out, int out_size, void* d_ws, size_t ws_size,
                              hipStream_t stream) {
  const float* x      = (const float*)d_in[0];        // (4, 50000, 128) f32
  const int*   eidx   = (const int*)d_in[1];          // (3, 350000) i32: src,dst,type
  const float* weight = (const float*)d_in[2];        // (1, 7, 128) f32
  const float* bias   = (const float*)d_in[3];        // (1, 1, 128) f32
  float*       out    = (float*)d_out;                // (4, 50000, 128) f32

  const int* src = eidx;
  const int* dst = eidx + N_EDGES;
  const int* typ = eidx + 2 * N_EDGES;

  // 1) out = bias (broadcast).  4*50000*128/4 = 6,400,000 float4 stores.
  const int n4 = BATCH * N_NODES * (DIM / 4);
  hex_init_out<<<(n4 + 255) / 256, 256, 0, stream>>>(out, bias, n4);

  // 2) scatter-add the weighted messages.  2048 blocks * 8 waves -> ~21 edges
  //    per wave via grid-stride.
  hex_edge_scatter<<<2048, 256, 0, stream>>>(x, src, dst, typ, weight, out, N_EDGES);
}